// Channel_attention_87660282512095
// MI455X (gfx1250) — compile-verified
//
#include <hip/hip_runtime.h>

// CDNA5 / gfx1250. wave32. WMMA f32 16x16x4 used as a vector-reduction engine.
typedef __attribute__((ext_vector_type(2))) float v2f;
typedef __attribute__((ext_vector_type(8))) float v8f;

#define HW        16384   // 128*128
#define NCH       256
#define NB        16
#define KSEL      16
#define NCHAN_ALL (NB * NCH)

// ---------------------------------------------------------------------------
// Pass 1: per-channel mean.  One block per (b,c) channel; 8 waves x 32 lanes.
// Each lane streams 16 float4 (b128) loads; every float4 feeds two
// V_WMMA_F32_16X16X4_F32 ops with B = ones, so D accumulates row-sums of the
// streamed data in fp32.  Two independent accumulator chains (acc0/acc1)
// break the per-WMMA RAW dependency on C so consecutive WMMAs dual-issue
// without s_delay_alu stalls; chains are merged once at the end.
// ---------------------------------------------------------------------------
__global__ __launch_bounds__(256) void mean_kernel(const float* __restrict__ x,
                                                   float* __restrict__ means) {
  const int ch  = blockIdx.x;                       // b*256 + c
  const int tid = threadIdx.x;
  const float4* __restrict__ src =
      (const float4*)(x + (size_t)ch * HW);

  v2f ones; ones[0] = 1.0f; ones[1] = 1.0f;
  v8f acc0 = {};
  v8f acc1 = {};

  // 256 threads * 16 iters * 4 floats = 16384 floats, fully coalesced.
  #pragma unroll
  for (int i = 0; i < 16; ++i) {
    float4 v = src[i * 256 + tid];
    v2f a0; a0[0] = v.x; a0[1] = v.y;
    v2f a1; a1[0] = v.z; a1[1] = v.w;
    // D[m,n] = sum_k A[m,k] * 1 + C[m,n]  -> 64-wide fp32 accumulate per op
    acc0 = __builtin_amdgcn_wmma_f32_16x16x4_f32(
        /*neg_a=*/false, a0, /*neg_b=*/false, ones,
        /*c_mod=*/(short)0, acc0, /*reuse_a=*/false, /*reuse_b=*/false);
    acc1 = __builtin_amdgcn_wmma_f32_16x16x4_f32(
        false, a1, false, ones, (short)0, acc1, false, false);
  }

  // Merge the two chains, then each lane sums its 8 accumulator VGPRs:
  // lanes 0-15 -> rowsums 0..7, lanes 16-31 -> rowsums 8..15 (columns are
  // replicas, so any column works).
  float s = (acc0[0] + acc1[0]) + (acc0[1] + acc1[1]) +
            (acc0[2] + acc1[2]) + (acc0[3] + acc1[3]) +
            (acc0[4] + acc1[4]) + (acc0[5] + acc1[5]) +
            (acc0[6] + acc1[6]) + (acc0[7] + acc1[7]);
  float wave_total = __shfl(s, 0, 32) + __shfl(s, 16, 32);

  __shared__ float wsum[8];
  const int wave = tid >> 5;
  if ((tid & 31) == 0) wsum[wave] = wave_total;
  __syncthreads();

  if (tid == 0) {
    float m = 0.0f;
    #pragma unroll
    for (int w = 0; w < 8; ++w) m += wsum[w];   // fixed order -> deterministic
    means[ch] = m * (1.0f / (float)HW);
  }
}

// ---------------------------------------------------------------------------
// Pass 2: bottom-16 selection per batch.  Thread t computes the exact stable
// ascending-argsort rank of channel t; ranks are a permutation of 0..255, so
// the 16 output slots are each written exactly once.
// ---------------------------------------------------------------------------
__global__ __launch_bounds__(256) void rank_kernel(const float* __restrict__ means,
                                                   int* __restrict__ sel) {
  const int b = blockIdx.x;
  const int t = threadIdx.x;
  __shared__ float m[NCH];
  m[t] = means[b * NCH + t];
  __syncthreads();

  const float mv = m[t];
  int rank = 0;
  #pragma unroll 8
  for (int c = 0; c < NCH; ++c) {
    float o = m[c];
    rank += (o < mv) || (o == mv && c < t);   // stable tie-break on index
  }
  if (rank < KSEL) sel[b * KSEL + rank] = t;
}

// ---------------------------------------------------------------------------
// Pass 3: gather selected channels.  grid = (16 chunks, 256 (b,k) pairs);
// each thread moves one float4 -> b128 load + b128 store, fully coalesced.
// ---------------------------------------------------------------------------
__global__ __launch_bounds__(256) void gather_kernel(const float* __restrict__ x,
                                                     const int* __restrict__ sel,
                                                     float* __restrict__ out) {
  const int bk = blockIdx.y;                 // b*16 + k
  const int b  = bk >> 4;
  const int ch = sel[bk];                    // uniform scalar load per block

  const float4* __restrict__ src =
      (const float4*)(x + ((size_t)b * NCH + ch) * HW);
  float4* __restrict__ dst = (float4*)(out + (size_t)bk * HW);

  const int e = blockIdx.x * 256 + threadIdx.x;   // 0..4095 float4s
  dst[e] = src[e];
}

// ---------------------------------------------------------------------------
extern "C" void kernel_launch(void* const* d_in, const int* in_sizes, int n_in,
                              void* d_out, int out_size, void* d_ws, size_t ws_size,
                              hipStream_t stream) {
  const float* x   = (const float*)d_in[0];
  float*       out = (float*)d_out;

  float* means = (float*)d_ws;                                    // 4096 f32
  int*   sel   = (int*)((char*)d_ws + NCHAN_ALL * sizeof(float)); // 256 i32

  mean_kernel  <<<NCHAN_ALL, 256, 0, stream>>>(x, means);
  rank_kernel  <<<NB,        256, 0, stream>>>(means, sel);
  gather_kernel<<<dim3(HW / (256 * 4), NB * KSEL), 256, 0, stream>>>(x, sel, out);
}